// PhiRestraint_29231547417079
// MI455X (gfx1250) — compile-verified
//
#include <hip/hip_runtime.h>
#include <cmath>

// ---------------------------------------------------------------------------
// PhiRestraint: angle-spline energy sum.
//   kernel 1: per (batch, pair-chunk) block -> stage CA[b],CB[b] to LDS via
//             async global->LDS loads, compute phi + cubic spline, reduce
//             block partial with V_WMMA_F32_16X16X4_F32, write to d_ws.
//   kernel 2: single block reduces all partials (fixed order) -> d_out[0].
// ---------------------------------------------------------------------------

typedef __attribute__((ext_vector_type(2))) float v2f;
typedef __attribute__((ext_vector_type(8))) float v8f;

#define AS_LOCAL __attribute__((address_space(3)))

// Wave32 sum reduction using the FP32 WMMA unit.
// A (16x4): VGPR0 = lane partial, VGPR1 = 0  => A[m][{0,2}] = p_m, p_{m+16}
// B (4x16): all ones => D[m][n] = p_m + p_{m+16} for every n.
// Sum D's 8 row-VGPRs per lane, then add lane^16 partner -> full wave sum,
// in a fixed deterministic order.
__device__ __forceinline__ float wave_reduce_sum(float v) {
  v2f a; a[0] = v;    a[1] = 0.0f;
  v2f b; b[0] = 1.0f; b[1] = 1.0f;
  v8f c = {};
  v8f d = __builtin_amdgcn_wmma_f32_16x16x4_f32(
      /*neg_a=*/false, a, /*neg_b=*/false, b,
      /*c_mod=*/(short)0, c, /*reuse_a=*/false, /*reuse_b=*/false);
  float s = ((d[0] + d[1]) + (d[2] + d[3])) + ((d[4] + d[5]) + (d[6] + d[7]));
  s += __shfl_xor(s, 16, 32);
  return s;
}

__device__ __forceinline__ void async_load16_to_lds(float* lds_dst,
                                                    const float* gbase,
                                                    unsigned byte_off) {
  // LDS byte offset: ptrtoint of an addrspace(3) pointer is the 32-bit offset.
  unsigned lds_off = (unsigned)(size_t)(AS_LOCAL float*)lds_dst;
  // GVS mode: mem_addr = SGPR_base + VGPR_offset (+0). 16B per lane -> LDS.
  asm volatile("global_load_async_to_lds_b128 %0, %1, %2"
               :: "v"(lds_off), "v"(byte_off), "s"(gbase)
               : "memory");
}

__device__ __forceinline__ void wait_async_lds() {
#if __has_builtin(__builtin_amdgcn_s_wait_asynccnt)
  __builtin_amdgcn_s_wait_asynccnt(0);
#else
  asm volatile("s_wait_asynccnt 0" ::: "memory");
#endif
}

__global__ __launch_bounds__(256)
void phi_pairs_kernel(const float* __restrict__ CA,      // [B, L, 3]
                      const float* __restrict__ CB,      // [B, L, 3]
                      const float* __restrict__ coeff,   // [L, L, K-1, 4]
                      const float* __restrict__ cutoffs, // [K]  (K == 16)
                      const int*   __restrict__ x_idx,   // [P]
                      const int*   __restrict__ y_idx,   // [P]
                      float*       __restrict__ partials,
                      int P, int L) {
  __shared__ float sCA[512 * 3];
  __shared__ float sCB[512 * 3];
  __shared__ float scut[16];
  __shared__ float swave[8];

  const int tid = threadIdx.x;
  const int b   = blockIdx.y;

  // ---- stage this batch's coordinates into LDS (async DMA path) ----
  const float* gCA = CA + (size_t)b * (size_t)(L * 3);
  const float* gCB = CB + (size_t)b * (size_t)(L * 3);
  int n4 = (L * 3) >> 2;                 // number of float4 chunks (384)
  if (n4 > 384) n4 = 384;                // LDS safety clamp
  for (int i = tid; i < n4; i += (int)blockDim.x) {
    const unsigned off = (unsigned)(i * 16);
    async_load16_to_lds(&sCA[i * 4], gCA, off);
    async_load16_to_lds(&sCB[i * 4], gCB, off);
  }
  if (tid < 16) scut[tid] = cutoffs[tid];
  wait_async_lds();
  __syncthreads();

  // Cutoffs into (uniform -> SGPR) registers for the branchless searchsorted.
  float cut[16];
#pragma unroll
  for (int k = 0; k < 16; ++k) cut[k] = cutoffs[k];

  const float eps = 1e-6f;
  float acc = 0.0f;

  for (int p = (int)(blockIdx.x * blockDim.x) + tid; p < P;
       p += (int)(gridDim.x * blockDim.x)) {
    const int xi = x_idx[p];
    const int yi = y_idx[p];

    const float xCAx = sCA[xi * 3 + 0], xCAy = sCA[xi * 3 + 1], xCAz = sCA[xi * 3 + 2];
    const float xCBx = sCB[xi * 3 + 0], xCBy = sCB[xi * 3 + 1], xCBz = sCB[xi * 3 + 2];
    const float yCBx = sCB[yi * 3 + 0], yCBy = sCB[yi * 3 + 1], yCBz = sCB[yi * 3 + 2];

    const float vx0 = xCAx - xCBx, vx1 = xCAy - xCBy, vx2 = xCAz - xCBz;
    const float vy0 = yCBx - xCBx, vy1 = yCBy - xCBy, vy2 = yCBz - xCBz;

    const float nx = sqrtf(vx0 * vx0 + vx1 * vx1 + vx2 * vx2);
    const float ny = sqrtf(vy0 * vy0 + vy1 * vy1 + vy2 * vy2);
    const bool valid = (nx > eps) && (ny > eps);
    const float denom = valid ? (nx * ny) : 1.0f;
    const float cang = (vx0 * vy0 + vx1 * vy1 + vx2 * vy2) / denom;
    const bool good = (1.0f - cang * cang) > eps;
    const bool msk = valid && good;
    const float cs = msk ? fminf(fmaxf(cang, -1.0f + 1e-7f), 1.0f - 1e-7f) : 0.0f;
    const float phi = acosf(cs);

    // searchsorted(cutoffs, phi, 'left') = count of cutoffs[k] < phi
    int cnt = 0;
#pragma unroll
    for (int k = 0; k < 16; ++k) cnt += (cut[k] < phi) ? 1 : 0;
    int idx = cnt - 1;
    idx = idx < 0 ? 0 : (idx > 14 ? 14 : idx);
    const float dx = phi - scut[idx];

    // one aligned 16B coefficient record (instead of the full 15x4 row)
    const size_t rec = ((size_t)xi * (size_t)L + (size_t)yi) * 15u + (size_t)idx;
    const float4 cf = *reinterpret_cast<const float4*>(coeff + rec * 4u);
    const float val = cf.w + dx * (cf.z + dx * (cf.y + dx * cf.x));

    acc += msk ? val : 0.0f;
  }

  // ---- deterministic block reduction: WMMA wave sums -> LDS -> thread 0 ----
  const float wsum = wave_reduce_sum(acc);
  if ((tid & 31) == 0) swave[tid >> 5] = wsum;
  __syncthreads();
  if (tid == 0) {
    float s = 0.0f;
#pragma unroll
    for (int w = 0; w < 8; ++w) s += swave[w];
    partials[(size_t)b * gridDim.x + blockIdx.x] = s;
  }
}

__global__ __launch_bounds__(256)
void reduce_partials_kernel(const float* __restrict__ partials,
                            float* __restrict__ out, int n) {
  __shared__ float swave[8];
  const int tid = threadIdx.x;
  float acc = 0.0f;
  for (int i = tid; i < n; i += 256) acc += partials[i];
  const float wsum = wave_reduce_sum(acc);
  if ((tid & 31) == 0) swave[tid >> 5] = wsum;
  __syncthreads();
  if (tid == 0) {
    float s = 0.0f;
#pragma unroll
    for (int w = 0; w < 8; ++w) s += swave[w];
    out[0] = s;
  }
}

extern "C" void kernel_launch(void* const* d_in, const int* in_sizes, int n_in,
                              void* d_out, int out_size, void* d_ws, size_t ws_size,
                              hipStream_t stream) {
  const float* CA      = (const float*)d_in[0];
  const float* CB      = (const float*)d_in[1];
  const float* coeff   = (const float*)d_in[2];
  const float* cutoffs = (const float*)d_in[3];
  const int*   x_idx   = (const int*)d_in[4];
  const int*   y_idx   = (const int*)d_in[5];

  const int K = in_sizes[3];               // 16
  const int P = in_sizes[4];               // 80000
  const double l2 = (double)in_sizes[2] / (4.0 * (double)(K - 1));
  const int L = (int)(sqrt(l2) + 0.5);     // 512
  const int B = in_sizes[0] / (3 * L);     // 8

  const int threads = 256;
  const int blocksX = 128;                 // 128 x 8 = 1024 blocks total
  float* partials = (float*)d_ws;          // 1024 floats of scratch

  dim3 grid((unsigned)blocksX, (unsigned)B, 1);
  phi_pairs_kernel<<<grid, threads, 0, stream>>>(CA, CB, coeff, cutoffs,
                                                 x_idx, y_idx, partials, P, L);
  reduce_partials_kernel<<<1, threads, 0, stream>>>(partials, (float*)d_out,
                                                    blocksX * B);
}